// Layer_70411693850655
// MI455X (gfx1250) — compile-verified
//
#include <hip/hip_runtime.h>

#define N_NODES 50000
#define N_EDGES 500000

typedef __attribute__((ext_vector_type(2))) float v2f;
typedef __attribute__((ext_vector_type(8))) float v8f;

__device__ __forceinline__ v8f wmma_f32_16x16x4(v2f a, v2f b, v8f c) {
  // 8 args: (neg_a, A, neg_b, B, c_mod, C, reuse_a, reuse_b)
  return __builtin_amdgcn_wmma_f32_16x16x4_f32(
      false, a, false, b, (short)0, c, false, false);
}

// Hardware fp32 global atomic add (no CAS loop): prefer the explicit builtin,
// then HIP's unsafeAtomicAdd, then the relaxed agent-scope atomic.
__device__ __forceinline__ void hw_atomic_add_f32(float* p, float v) {
#if defined(__has_builtin) && __has_builtin(__builtin_amdgcn_global_atomic_fadd_f32)
  typedef float __attribute__((address_space(1))) gfloat;
  __builtin_amdgcn_global_atomic_fadd_f32((gfloat*)p, v);
#elif defined(__HIP_DEVICE_COMPILE__) && defined(__AMDGCN__)
  unsafeAtomicAdd(p, v);
#else
  __hip_atomic_fetch_add(p, v, __ATOMIC_RELAXED, __HIP_MEMORY_SCOPE_AGENT);
#endif
}

// ---------------------------------------------------------------- zero
__global__ void zero_f4(float4* __restrict__ p, unsigned n4) {
  unsigned i = blockIdx.x * blockDim.x + threadIdx.x;
  unsigned stride = gridDim.x * blockDim.x;
  float4 z = make_float4(0.f, 0.f, 0.f, 0.f);
  for (; i < n4; i += stride) p[i] = z;
}

// ---------------------------------------------------------------- edge scatter
// One wave32 per edge. Lane l covers elements (d = (l>>4)+2j, u = l&15),
// j = 0..7 -> 8 contiguous, coalesced global atomic f32 adds per lane.
__global__ void edge_scatter(const float* __restrict__ pos,
                             const float* __restrict__ ns,
                             const int* __restrict__ snd,
                             const int* __restrict__ rcv,
                             float* __restrict__ agg,
                             int aggStride) {
  unsigned tid = blockIdx.x * blockDim.x + threadIdx.x;
  unsigned e = tid >> 5;
  if (e >= N_EDGES) return;
  const int lane = threadIdx.x & 31;

  int s = snd[e];
  int r = rcv[e];
  float dx = pos[3 * r + 0] - pos[3 * s + 0];
  float dy = pos[3 * r + 1] - pos[3 * s + 1];
  float dz = pos[3 * r + 2] - pos[3 * s + 2];
  float rn = rsqrtf(dx * dx + dy * dy + dz * dz);
  float x = dx * rn, y = dy * rn, z = dz * rn;
  float x2 = x * x, y2 = y * y, z2 = z * z;

  const float S3 = 1.7320508075688772f;
  const float S5 = 2.23606797749979f;
  const float S7 = 2.6457513110645906f;
  const float C42 = 1.0801234497346435f;   // sqrt(42)/6
  const float C168 = 1.620185174601965f;   // sqrt(168)/8
  const float HS7 = 1.3228756555322954f;   // sqrt(7)/2

  float ev[16];
  ev[0] = 1.0f;
  ev[1] = S3 * x; ev[2] = S3 * y; ev[3] = S3 * z;
  float a0 = S3 * x * z;
  float a1 = S3 * x * y;
  float a2 = y2 - 0.5f * (x2 + z2);
  float a3 = S3 * y * z;
  float a4 = 0.5f * S3 * (z2 - x2);
  ev[4] = S5 * a0; ev[5] = S5 * a1; ev[6] = S5 * a2; ev[7] = S5 * a3; ev[8] = S5 * a4;
  float q = 4.0f * y2 - x2 - z2;
  ev[9]  = S7 * (C42 * (a0 * z + a4 * x));
  ev[10] = S7 * (S7 * a0 * y);
  ev[11] = S7 * (C168 * q * x);
  ev[12] = S7 * (HS7 * y * (2.0f * y2 - 3.0f * x2 - 3.0f * z2));
  ev[13] = S7 * (C168 * z * q);
  ev[14] = S7 * (S7 * a4 * y);
  ev[15] = S7 * (C42 * (a4 * z - a0 * x));

  float su = ns[s * 16 + (lane & 15)];
  int hi = lane >> 4;
  float* base = agg + (size_t)r * (size_t)aggStride;
#pragma unroll
  for (int j = 0; j < 8; ++j) {
    float v = (hi ? ev[2 * j + 1] : ev[2 * j]) * su;
    hw_atomic_add_f32(base + lane + 32 * j, v);
  }
}

// ---------------------------------------------------------------- node GEMM (WMMA)
// Tall-skinny GEMM for irrep block: rows = flattened (node, d_local),
// D(16 x 64) = A(16 x 16) * W(16 x 64), via 4 K-steps of V_WMMA_F32_16X16X4_F32
// times 4 N-tiles. WITH_SC fuses node_scalars @ W_sc as 4 extra K-steps.
template <int DK, int OFF, bool WITH_SC>
__global__ void node_gemm(const float* __restrict__ agg, int aggStride,
                          const float* __restrict__ W,
                          const float* __restrict__ ns,
                          const float* __restrict__ Wsc,
                          float* __restrict__ out) {
  const int lane = threadIdx.x & 31;
  const int h = lane >> 4;
  const int col = lane & 15;
  const unsigned wave = (blockIdx.x * blockDim.x + threadIdx.x) >> 5;
  const unsigned nwaves = (gridDim.x * blockDim.x) >> 5;
  const unsigned ntiles = (unsigned)(N_NODES * DK) / 16u;

  const float SA = 0.25f / 10.0f;  // (1/sqrt(16)) / DENOM  (agg rows)
  const float SS = 0.25f;          // 1/sqrt(16)            (skip rows)

  // Preload scaled B fragments: B[k][n] frag = {W[k0, c], W[k0+1, c]}
  v2f B[4][4];
#pragma unroll
  for (int kt = 0; kt < 4; ++kt) {
    int k0 = 4 * kt + 2 * h;
#pragma unroll
    for (int nt = 0; nt < 4; ++nt) {
      B[kt][nt].x = W[k0 * 64 + nt * 16 + col] * SA;
      B[kt][nt].y = W[(k0 + 1) * 64 + nt * 16 + col] * SA;
    }
  }
  v2f Bs[4][4];
  if (WITH_SC) {
#pragma unroll
    for (int kt = 0; kt < 4; ++kt) {
      int k0 = 4 * kt + 2 * h;
#pragma unroll
      for (int nt = 0; nt < 4; ++nt) {
        Bs[kt][nt].x = Wsc[k0 * 64 + nt * 16 + col] * SS;
        Bs[kt][nt].y = Wsc[(k0 + 1) * 64 + nt * 16 + col] * SS;
      }
    }
  }

  for (unsigned t = wave; t < ntiles; t += nwaves) {
    // A-load row for this lane: global flat row R; both halves load row l&15.
    unsigned R = t * 16u + (unsigned)col;
    unsigned n = R / (unsigned)DK;
    unsigned d = (unsigned)OFF + (R - n * (unsigned)DK);
    const float* arow = agg + (size_t)n * (size_t)aggStride + d * 16u;

    v8f acc[4] = {};
#pragma unroll
    for (int kt = 0; kt < 4; ++kt) {
      v2f a;
      a.x = arow[4 * kt + 2 * h];
      a.y = arow[4 * kt + 2 * h + 1];
#pragma unroll
      for (int nt = 0; nt < 4; ++nt)
        acc[nt] = wmma_f32_16x16x4(a, B[kt][nt], acc[nt]);
    }
    if (WITH_SC) {
      const float* srow = ns + (size_t)n * 16u;  // DK==1 -> d is always OFF
      v2f a;
#pragma unroll
      for (int kt = 0; kt < 4; ++kt) {
        a.x = srow[4 * kt + 2 * h];
        a.y = srow[4 * kt + 2 * h + 1];
#pragma unroll
        for (int nt = 0; nt < 4; ++nt)
          acc[nt] = wmma_f32_16x16x4(a, Bs[kt][nt], acc[nt]);
      }
    }
    // Store: c[r] = D[r + 8h, col]; output row m -> (node, d) -> out row ptr.
#pragma unroll
    for (int r = 0; r < 8; ++r) {
      unsigned m = (unsigned)(r + 8 * h);
      unsigned Rm = t * 16u + m;
      unsigned nm = Rm / (unsigned)DK;
      unsigned dm = (unsigned)OFF + (Rm - nm * (unsigned)DK);
      float* orow = out + (size_t)nm * 1024u + dm * 64u;
#pragma unroll
      for (int nt = 0; nt < 4; ++nt)
        orow[nt * 16 + col] = acc[nt][r];
    }
  }
}

// ---------------------------------------------------------------- host
extern "C" void kernel_launch(void* const* d_in, const int* in_sizes, int n_in,
                              void* d_out, int out_size, void* d_ws, size_t ws_size,
                              hipStream_t stream) {
  (void)in_sizes; (void)n_in; (void)out_size;
  const float* positions = (const float*)d_in[0];
  const float* node_scalars = (const float*)d_in[1];
  const float* W0 = (const float*)d_in[2];
  const float* W1 = (const float*)d_in[3];
  const float* W2 = (const float*)d_in[4];
  const float* W3 = (const float*)d_in[5];
  const float* Wsc = (const float*)d_in[6];
  const int* senders = (const int*)d_in[7];
  const int* receivers = (const int*)d_in[8];
  float* out = (float*)d_out;

  const int TB = 256;
  size_t need = (size_t)N_NODES * 256u * sizeof(float);
  bool use_ws = (ws_size >= need);
  float* agg = use_ws ? (float*)d_ws : out;
  int aggStride = use_ws ? 256 : 1024;

  // 1) zero the accumulator region (whole out if it doubles as scratch)
  {
    unsigned n4 = (use_ws ? (unsigned)(N_NODES * 256u)
                          : (unsigned)(N_NODES * 1024u)) / 4u;
    unsigned blocks = (n4 + TB - 1) / TB;
    if (blocks > 4096u) blocks = 4096u;
    zero_f4<<<dim3(blocks), dim3(TB), 0, stream>>>((float4*)agg, n4);
  }

  // 2) edge scatter: one wave per edge
  {
    unsigned blocks = (unsigned)(((size_t)N_EDGES * 32 + TB - 1) / TB);
    edge_scatter<<<dim3(blocks), dim3(TB), 0, stream>>>(
        positions, node_scalars, senders, receivers, agg, aggStride);
  }

  // 3) per-irrep WMMA GEMMs
  auto blocks_for = [&](int dk) -> unsigned {
    unsigned tiles = (unsigned)(N_NODES * dk) / 16u;
    return (tiles + 7u) / 8u;  // 8 waves per 256-thread block
  };
  if (use_ws) {
    node_gemm<1, 0, true><<<dim3(blocks_for(1)), dim3(TB), 0, stream>>>(
        agg, aggStride, W0, node_scalars, Wsc, out);
    node_gemm<3, 1, false><<<dim3(blocks_for(3)), dim3(TB), 0, stream>>>(
        agg, aggStride, W1, nullptr, nullptr, out);
    node_gemm<5, 4, false><<<dim3(blocks_for(5)), dim3(TB), 0, stream>>>(
        agg, aggStride, W2, nullptr, nullptr, out);
    node_gemm<7, 9, false><<<dim3(blocks_for(7)), dim3(TB), 0, stream>>>(
        agg, aggStride, W3, nullptr, nullptr, out);
  } else {
    // out[:,0:256] doubles as agg: order chosen so every GEMM reads its A
    // rows before any later kernel overwrites them.
    node_gemm<5, 4, false><<<dim3(blocks_for(5)), dim3(TB), 0, stream>>>(
        agg, aggStride, W2, nullptr, nullptr, out);
    node_gemm<7, 9, false><<<dim3(blocks_for(7)), dim3(TB), 0, stream>>>(
        agg, aggStride, W3, nullptr, nullptr, out);
    node_gemm<3, 1, false><<<dim3(blocks_for(3)), dim3(TB), 0, stream>>>(
        agg, aggStride, W1, nullptr, nullptr, out);
    node_gemm<1, 0, true><<<dim3(blocks_for(1)), dim3(TB), 0, stream>>>(
        agg, aggStride, W0, node_scalars, Wsc, out);
  }
}